// Domain_Transform_48301202211290
// MI455X (gfx1250) — compile-verified
//
#include <hip/hip_runtime.h>
#include <math.h>

// Domain-transform recursive filter, CDNA5 (gfx1250).
// Chunked-scan formulation: each 16-wide segment of the recurrence
//   x_i = a_i x_{i-1} + (1-a_i) img_i
// becomes   x_tile = T(16x16, lower-tri) * b_tile(16x16ch) + P * carry,
// with T built in log2-space from a lane prefix-sum and the 16x16 product
// executed as 4 chained V_WMMA_F32_16X16X4_F32 (fp32 tensor path).

typedef __attribute__((ext_vector_type(2))) float v2f;
typedef __attribute__((ext_vector_type(8))) float v8f;

#define WIDTH  512
#define SEGS   32
#define XPITCH 513      // LDS pitch (floats) per channel row: 513 % 64 == 1 -> conflict-free

__device__ __forceinline__ float exp2_fast(float x) {
    return __builtin_amdgcn_exp2f(x);
}

// One scan direction over the 16(ch) x 512(pos) tile resident in LDS.
// bwd=false: x_i = a_i x_{i-1} + omF_i * X_i          (left -> right)
// bwd=true : y uses shifted coeffs a'_r = a_{512-r}, data reversed in place.
__device__ __forceinline__ void scan_dir(float* X, const float* tF, const float* omF,
                                         int lid, int half, int khi, bool bwd)
{
    float carry = 0.0f;
    for (int s = 0; s < SEGS; ++s) {
        const int base = s * 16;
        const int idx  = base + half;

        // per-lane log2(a) for this segment position
        float t;
        if (!bwd) t = tF[idx];
        else      t = (idx == 0) ? -512.0f : tF[WIDTH - idx];

        // inclusive 16-lane prefix sum (both half-waves hold identical S)
        float S = t;
#pragma unroll
        for (int off = 1; off < 16; off <<= 1) {
            float o = __shfl(S, (lid - off) & 31);
            if (half >= off) S += o;
        }
        const float expS = exp2_fast(S);     // P[p] = prod_{k<=p} a_k

        v8f acc = {0.f,0.f,0.f,0.f,0.f,0.f,0.f,0.f};
#pragma unroll
        for (int kk = 0; kk < 4; ++kk) {
            const int m0 = kk * 4 + khi * 2;
            // T[p][m] = exp2(S_p - S_m)  (m<p), 1 (m==p), 0 (m>p)
            const float Sm0 = __shfl(S, m0);
            const float Sm1 = __shfl(S, m0 + 1);
            v2f Af;
            Af.x = (m0     < half) ? exp2_fast(S - Sm0) : ((m0     == half) ? 1.0f : 0.0f);
            Af.y = (m0 + 1 < half) ? exp2_fast(S - Sm1) : ((m0 + 1 == half) ? 1.0f : 0.0f);

            // B fragment: rows m0,m0+1 of b[m][chan], chan = half
            const int p0 = base + m0;
            v2f Bf;
            if (!bwd) {
                Bf.x = omF[p0]     * X[half * XPITCH + p0];
                Bf.y = omF[p0 + 1] * X[half * XPITCH + p0 + 1];
            } else {
                const float om0 = (p0 == 0) ? 1.0f : omF[WIDTH - p0];
                const float om1 = omF[WIDTH - (p0 + 1)];
                Bf.x = om0 * X[half * XPITCH + (WIDTH - 1 - p0)];
                Bf.y = om1 * X[half * XPITCH + (WIDTH - 1 - (p0 + 1))];
            }

            acc = __builtin_amdgcn_wmma_f32_16x16x4_f32(
                      false, Af, false, Bf, (short)0, acc, false, false);
        }

        // rank-1 carry term: acc[j][c] += P[j] * carry[c]
#pragma unroll
        for (int rr = 0; rr < 8; ++rr) {
            const float Pj = __shfl(expS, rr + 8 * khi);
            acc[rr] += Pj * carry;
        }
        // next carry = row j==15 (lives in VGPR 7 of lanes 16..31), chan = half
        carry = __shfl(acc[7], 16 + half);

        // write tile back to LDS (in place)
#pragma unroll
        for (int rr = 0; rr < 8; ++rr) {
            const int j = base + rr + 8 * khi;
            const int w = bwd ? (WIDTH - 1 - j) : j;
            X[half * XPITCH + w] = acc[rr];
        }
    }
}

__global__ __launch_bounds__(32)
void dt_pass_kernel(const float* in, float* out, const float* edge,
                    float clog2, int vertical)
{
    extern __shared__ float smem[];
    float* X   = smem;                    // [16][XPITCH]
    float* tF  = smem + 16 * XPITCH;      // [512]  log2(a_w)
    float* omF = tF + WIDTH;              // [512]  1 - a_w

    const int lid  = threadIdx.x;         // 0..31 (wave32)
    const int half = lid & 15;
    const int khi  = lid >> 4;

    const int nb  = blockIdx.x;           // nRows*2 blocks
    const int cg  = nb & 1;               // channel group (0: ch0-15, 1: ch16-31)
    const int row = nb >> 1;              // 0..2047
    const int b   = row >> 9;
    const int r   = row & 511;

    const int C = 32, H = 512, Wd = 512;
    const long long chStride = (long long)H * Wd;
    long long imgBase, edgeBase;
    int posStride, eStride;
    if (!vertical) {
        imgBase  = (long long)b * C * H * Wd + (long long)r * Wd;
        posStride = 1;
        edgeBase = (long long)b * H * Wd + (long long)r * Wd;
        eStride  = 1;
    } else {
        imgBase  = (long long)b * C * H * Wd + r;
        posStride = Wd;
        edgeBase = (long long)b * H * Wd + r;
        eStride  = Wd;
    }
    imgBase += (long long)(cg * 16) * chStride;

    // ---- stage 16 channels x 512 positions into LDS ----
    for (int i = lid; i < 16 * WIDTH; i += 32) {
        const int ch = i >> 9, w = i & (WIDTH - 1);
        X[ch * XPITCH + w] = in[imgBase + (long long)ch * chStride + (long long)w * posStride];
    }
    // ---- per-row coefficients: t = log2(a) = clog2 * D,  om = 1 - a ----
    for (int w = lid; w < WIDTH; w += 32) {
        const float Dv = 1.0f + 150.0f * edge[edgeBase + (long long)w * eStride];
        float t = clog2 * Dv;
        if (w == 0) t = -512.0f;          // force a_0 = 0  (x_0 = img_0)
        tF[w]  = t;
        omF[w] = 1.0f - exp2_fast(t);
    }
    __syncthreads();

    scan_dir(X, tF, omF, lid, half, khi, false);   // forward
    scan_dir(X, tF, omF, lid, half, khi, true);    // backward

    __syncthreads();
    // ---- write back ----
    for (int i = lid; i < 16 * WIDTH; i += 32) {
        const int ch = i >> 9, w = i & (WIDTH - 1);
        out[imgBase + (long long)ch * chStride + (long long)w * posStride] = X[ch * XPITCH + w];
    }
}

extern "C" void kernel_launch(void* const* d_in, const int* in_sizes, int n_in,
                              void* d_out, int out_size, void* d_ws, size_t ws_size,
                              hipStream_t stream) {
    (void)in_sizes; (void)n_in; (void)out_size; (void)d_ws; (void)ws_size;
    const float* img  = (const float*)d_in[0];
    const float* edge = (const float*)d_in[1];
    float* out = (float*)d_out;

    const double SQRT2 = 1.4142135623730951;
    const double SQRT3 = 1.7320508075688772;
    const double LOG2E = 1.4426950408889634;
    const double denom = 3.872983346207417;     // sqrt(4^2 - 1)
    const double sigma1 = 60.0 * SQRT3 * 2.0 / denom;
    const double sigma2 = 60.0 * SQRT3 * 1.0 / denom;
    const float c1 = (float)(-SQRT2 / sigma1 * LOG2E);  // log2-space coefficient
    const float c2 = (float)(-SQRT2 / sigma2 * LOG2E);

    const dim3 grid(4096), block(32);           // 2048 rows x 2 channel groups
    const size_t shmem = (size_t)(16 * XPITCH + 2 * WIDTH) * sizeof(float);

    // iteration 1: H then V (in-place after first pass; block working sets are disjoint)
    dt_pass_kernel<<<grid, block, shmem, stream>>>(img, out, edge, c1, 0);
    dt_pass_kernel<<<grid, block, shmem, stream>>>(out, out, edge, c1, 1);
    // iteration 2
    dt_pass_kernel<<<grid, block, shmem, stream>>>(out, out, edge, c2, 0);
    dt_pass_kernel<<<grid, block, shmem, stream>>>(out, out, edge, c2, 1);
}